// LBPKernel_28638841930409
// MI455X (gfx1250) — compile-verified
//
#include <hip/hip_runtime.h>
#include <math.h>

#define B_  16
#define H_  1024
#define W_  1024
#define TX  128          // tile width  (interior)
#define TY  64           // tile height (interior)
#define S_  (TX + 4)     // LDS row stride (132) — avoids systematic bank alignment
#define NWAVE 8          // 256 threads / wave32

typedef float v2f __attribute__((ext_vector_type(2)));
typedef float v8f __attribute__((ext_vector_type(8)));

// ---------------------------------------------------------------------------
// Kernel 0: zero the global histogram in workspace (deterministic every call)
// ---------------------------------------------------------------------------
__global__ void zero_hist_kernel(unsigned* __restrict__ ghist) {
    ghist[threadIdx.x] = 0u;
}

// ---------------------------------------------------------------------------
// Kernel 1: rgb->gray, 3x3 LBP codes, LDS-privatized 256-bin histogram
// ---------------------------------------------------------------------------
__global__ __launch_bounds__(256) void lbp_hist_kernel(const float* __restrict__ img,
                                                       unsigned* __restrict__ ghist) {
    __shared__ float    tile[(TY + 2) * S_];        // gray tile with 1-px halo
    __shared__ unsigned shist[NWAVE * 256];         // per-wave histogram copies

    const int tid = threadIdx.x;
    const int bx = blockIdx.x, by = blockIdx.y, b = blockIdx.z;

    for (int i = tid; i < NWAVE * 256; i += 256) shist[i] = 0u;

    const int x0 = bx * TX - 1;
    const int y0 = by * TY - 1;
    const float* rp = img + (size_t)(b * 3 + 0) * H_ * W_;
    const float* gp = img + (size_t)(b * 3 + 1) * H_ * W_;
    const float* bp = img + (size_t)(b * 3 + 2) * H_ * W_;

    // Load halo tile: gray = 0.299R + 0.587G + 0.114B, zero outside image
    const int HALO_W = TX + 2;
    const int HALO_N = (TY + 2) * HALO_W;
    for (int i = tid; i < HALO_N; i += 256) {
        int ly = i / HALO_W;
        int lx = i - ly * HALO_W;
        int gy = y0 + ly;
        int gx = x0 + lx;
        float g = 0.0f;
        if (gy >= 0 && gy < H_ && gx >= 0 && gx < W_) {
            size_t off = (size_t)gy * W_ + (size_t)gx;
            g = rp[off] * 0.299f + gp[off] * 0.587f + bp[off] * 0.114f;
        }
        tile[ly * S_ + lx] = g;
    }
    __syncthreads();

    // LBP codes: bit i set if (neighbor_i - center) >= 0; weights 2^0..2^7.
    // Neighbor order matches the filter bank: TR, R, BR, B, BL, L, TL, T.
    unsigned* myhist = &shist[(tid >> 5) * 256];
    for (int i = tid; i < TX * TY; i += 256) {
        int lx = i & (TX - 1);
        int ly = i >> 7;                    // TX == 128
        int p  = (ly + 1) * S_ + (lx + 1);
        float c = tile[p];
        unsigned code = 0u;
        code |= (tile[p - S_ + 1] >= c) ? 1u   : 0u;   // top-right
        code |= (tile[p      + 1] >= c) ? 2u   : 0u;   // right
        code |= (tile[p + S_ + 1] >= c) ? 4u   : 0u;   // bottom-right
        code |= (tile[p + S_    ] >= c) ? 8u   : 0u;   // bottom
        code |= (tile[p + S_ - 1] >= c) ? 16u  : 0u;   // bottom-left
        code |= (tile[p      - 1] >= c) ? 32u  : 0u;   // left
        code |= (tile[p - S_ - 1] >= c) ? 64u  : 0u;   // top-left
        code |= (tile[p - S_    ] >= c) ? 128u : 0u;   // top
        atomicAdd(&myhist[code], 1u);
    }
    __syncthreads();

    // Reduce the 8 wave-private copies; one global atomic per bin per block
    unsigned s = 0u;
    #pragma unroll
    for (int w = 0; w < NWAVE; ++w) s += shist[w * 256 + tid];
    if (s) atomicAdd(&ghist[tid], s);
}

// ---------------------------------------------------------------------------
// Kernel 2: standardize histogram. Sum and sum-of-squares of the 256 bins are
// computed with V_WMMA_F32_16X16X4_F32: A = hist (resp. hist^2) as a 16x16
// matrix fed in 4 chunks of K=4, B = ones (layout-independent), D accumulates
// row sums. One wave32, EXEC all ones (WMMA requirement).
// ---------------------------------------------------------------------------
__global__ __launch_bounds__(32) void finalize_kernel(const unsigned* __restrict__ ghist,
                                                      float* __restrict__ out) {
    const int lane = threadIdx.x;           // 0..31
    const int m    = lane & 15;             // A-matrix row (lanes 0-15 / 16-31)
    const int khi  = (lane >> 4) * 2;       // VGPR0 holds K=0 (lo lanes) / K=2 (hi)

    v2f ones; ones[0] = 1.0f; ones[1] = 1.0f;
    v8f s1 = {};   // row-sums of H
    v8f s2 = {};   // row-sums of H^2

    #pragma unroll
    for (int c = 0; c < 4; ++c) {
        int base = m * 16 + c * 4 + khi;
        float a0 = (float)ghist[base];
        float a1 = (float)ghist[base + 1];
        v2f av;  av[0]  = a0;      av[1]  = a1;
        v2f av2; av2[0] = a0 * a0; av2[1] = a1 * a1;
        s1 = __builtin_amdgcn_wmma_f32_16x16x4_f32(false, av,  false, ones,
                                                   (short)0, s1, false, false);
        s2 = __builtin_amdgcn_wmma_f32_16x16x4_f32(false, av2, false, ones,
                                                   (short)0, s2, false, false);
    }

    // D layout: lane holds rows {v + 8*(lane>=16)} in VGPR v (col = lane&15).
    float p1 = 0.0f, p2 = 0.0f;
    #pragma unroll
    for (int v = 0; v < 8; ++v) { p1 += s1[v]; p2 += s2[v]; }
    p1 += __shfl_xor(p1, 16, 32);           // combine row halves -> full sums
    p2 += __shfl_xor(p2, 16, 32);

    float mean = p1 * (1.0f / 256.0f);
    float var  = (p2 - 256.0f * mean * mean) * (1.0f / 255.0f);  // ddof = 1
    float inv  = 1.0f / sqrtf(var);

    #pragma unroll
    for (int j = 0; j < 8; ++j) {
        int idx = lane + 32 * j;
        out[idx] = ((float)ghist[idx] - mean) * inv;
    }
}

// ---------------------------------------------------------------------------
extern "C" void kernel_launch(void* const* d_in, const int* in_sizes, int n_in,
                              void* d_out, int out_size, void* d_ws, size_t ws_size,
                              hipStream_t stream) {
    const float* img   = (const float*)d_in[0];   // [16,3,1024,1024] fp32
    unsigned*    ghist = (unsigned*)d_ws;         // 256 u32 bins
    float*       out   = (float*)d_out;           // [1,256] fp32

    zero_hist_kernel<<<1, 256, 0, stream>>>(ghist);

    dim3 grid(W_ / TX, H_ / TY, B_);              // 8 x 16 x 16 = 2048 blocks
    lbp_hist_kernel<<<grid, 256, 0, stream>>>(img, ghist);

    finalize_kernel<<<1, 32, 0, stream>>>(ghist, out);
}